// NGCF_80006650790305
// MI455X (gfx1250) — compile-verified
//
#include <hip/hip_runtime.h>
#include <math.h>

#define N_USER 100000
#define N_ITEM 50000
#define NNODE  150000
#define DIM    64
#define OUTW   256   // 4 * DIM

typedef __attribute__((ext_vector_type(2))) float v2f;
typedef __attribute__((ext_vector_type(4))) float v4f;
typedef __attribute__((ext_vector_type(8))) float v8f;

// ---------------------------------------------------------------------------
// Kernel 1: E = concat(user_emb, item_emb); out[:, 0:64] = E  (slice 0)
// out is write-once/never re-read -> nontemporal stores keep L2 for E/front.
// ---------------------------------------------------------------------------
__global__ void init_concat(const float* __restrict__ user,
                            const float* __restrict__ item,
                            float* __restrict__ E,
                            float* __restrict__ out)
{
    int idx = blockIdx.x * blockDim.x + threadIdx.x;   // one float4 per thread
    if (idx >= NNODE * (DIM / 4)) return;
    int row = idx / (DIM / 4);
    int c4  = (idx % (DIM / 4)) * 4;
    v4f v = (row < N_USER)
          ? *(const v4f*)(user + (size_t)row * DIM + c4)
          : *(const v4f*)(item + (size_t)(row - N_USER) * DIM + c4);
    *(v4f*)(E + (size_t)row * DIM + c4) = v;
    __builtin_nontemporal_store(v, (v4f*)(out + (size_t)row * OUTW + c4));
}

// ---------------------------------------------------------------------------
// Kernel 2: COO SpMM scatter: front[row] += val * E[col]
// One wave32 per 32-edge chunk (grid-stride). Edge triples are staged into
// the wave's private LDS slab with async global->LDS loads, double buffered
// (ASYNCcnt pipeline): chunk k+1 streams in while chunk k is processed.
// Lane l owns columns {2l, 2l+1}; scatter uses native device-scope
// global_atomic_add_f32. The atomic asm deliberately has NO "memory"
// clobber so the scheduler can pipeline the data-dependent E[col] gathers
// across edges; the async/wait asm keeps the clobber to fence the LDS
// double buffer correctly.
// ---------------------------------------------------------------------------
__global__ void spmm_scatter(const int*   __restrict__ hrow,
                             const int*   __restrict__ hcol,
                             const float* __restrict__ hval,
                             const float* __restrict__ E,
                             float*       __restrict__ front,
                             int nnz)
{
    __shared__ int   s_row[8][64];   // [wave in block][2 buffers x 32 edges]
    __shared__ int   s_col[8][64];
    __shared__ float s_val[8][64];

    const int lane    = threadIdx.x & 31;
    const int wid     = threadIdx.x >> 5;
    const int wave    = (blockIdx.x * blockDim.x + threadIdx.x) >> 5;
    const int nwaves  = (gridDim.x * blockDim.x) >> 5;
    const int nchunks = (nnz + 31) >> 5;
    if (wave >= nchunks) return;

    auto issue = [&](int ch, int buf) {
        int e = ch * 32 + lane;
        if (e > nnz - 1) e = nnz - 1;            // clamp: never read OOB
        unsigned lr = (unsigned)(size_t)&s_row[wid][buf * 32 + lane];
        unsigned lc = (unsigned)(size_t)&s_col[wid][buf * 32 + lane];
        unsigned lv = (unsigned)(size_t)&s_val[wid][buf * 32 + lane];
        asm volatile("global_load_async_to_lds_b32 %0, %1, off"
                     :: "v"(lr), "v"(hrow + e) : "memory");
        asm volatile("global_load_async_to_lds_b32 %0, %1, off"
                     :: "v"(lc), "v"(hcol + e) : "memory");
        asm volatile("global_load_async_to_lds_b32 %0, %1, off"
                     :: "v"(lv), "v"(hval + e) : "memory");
    };

    int buf = 0;
    issue(wave, buf);

    for (int ch = wave; ch < nchunks; ch += nwaves) {
        const int next = ch + nwaves;
        if (next < nchunks) {
            issue(next, buf ^ 1);                           // prefetch next chunk
            asm volatile("s_wait_asynccnt 3" ::: "memory"); // current buf ready
        } else {
            asm volatile("s_wait_asynccnt 0" ::: "memory");
        }

        const int base = ch * 32;
        const int cnt  = (nnz - base < 32) ? (nnz - base) : 32;
        const int bb   = buf * 32;

        for (int j = 0; j < cnt; ++j) {
            const int   rj = s_row[wid][bb + j];   // uniform ds_load broadcast
            const int   cj = s_col[wid][bb + j];
            const float wj = s_val[wid][bb + j];

            if (j + 2 < cnt) {                     // hide the gather latency
                int cp = s_col[wid][bb + j + 2];
                __builtin_prefetch(E + (size_t)cp * DIM + (size_t)(lane & 1) * 32, 0, 3);
            }

            const v2f ev = *(const v2f*)(E + (size_t)cj * DIM + 2 * lane);
            const float ax = wj * ev.x;
            const float ay = wj * ev.y;
            float* dst = front + (size_t)rj * DIM + 2 * lane;
            // fire-and-forget f32 adds at the L2 atomic units, device scope;
            // second add reuses the same address pair via offset:4
            asm volatile("global_atomic_add_f32 %0, %1, off scope:SCOPE_DEV"
                         :: "v"(dst), "v"(ax));
            asm volatile("global_atomic_add_f32 %0, %1, off offset:4 scope:SCOPE_DEV"
                         :: "v"(dst), "v"(ay));
        }
        buf ^= 1;
    }
}

// ---------------------------------------------------------------------------
// Kernel 3: fused  y = leaky(front*(1+E)); E = y;
//           out[:, slice*64:(slice+1)*64] = y / max(||y||_2, 1e-12)
// One wave32 per 16-row tile in the native V_WMMA_F32_16X16X4_F32 A layout:
//   lane l: row m = l&15, K-half kh = l>>4; chunk c covers cols {4c+2kh, +1}
// Row sums-of-squares accumulate on the matrix pipe via A = y*y, B = ones.
// ---------------------------------------------------------------------------
__global__ void fuse_norm(const float* __restrict__ front,
                          float*       __restrict__ E,
                          float*       __restrict__ out,
                          int slice)
{
    const int lane = threadIdx.x & 31;
    const int wid  = threadIdx.x >> 5;
    const int tile = blockIdx.x * (blockDim.x >> 5) + wid;
    if (tile >= NNODE / 16) return;

    const int m    = lane & 15;
    const int kh   = lane >> 4;
    const int rowg = tile * 16 + m;

    const float* fp = front + (size_t)rowg * DIM + 2 * kh;
    float*       ep = E     + (size_t)rowg * DIM + 2 * kh;

    v2f y[16];
    v8f acc = {0.f, 0.f, 0.f, 0.f, 0.f, 0.f, 0.f, 0.f};
    const v2f ones = {1.0f, 1.0f};

#pragma unroll
    for (int c = 0; c < 16; ++c) {
        v2f f = *(const v2f*)(fp + 4 * c);
        v2f e = *(const v2f*)(ep + 4 * c);
        v2f t;
        t.x = f.x * (1.0f + e.x);
        t.y = f.y * (1.0f + e.y);
        t.x = (t.x > 0.0f) ? t.x : 0.01f * t.x;   // leaky_relu, slope 0.01
        t.y = (t.y > 0.0f) ? t.y : 0.01f * t.y;
        y[c] = t;
        *(v2f*)(ep + 4 * c) = t;                   // E <- unnormalized activation

        v2f sq; sq.x = t.x * t.x; sq.y = t.y * t.y;
        acc = __builtin_amdgcn_wmma_f32_16x16x4_f32(
                  false, sq, false, ones, (short)0, acc, false, false);
    }

    // acc VGPR j: lanes 0-15 = rowsum(row j), lanes 16-31 = rowsum(row j+8),
    // replicated across N. Extract this lane's own row sum.
    const int k = lane & 7;
    float s0 = (k & 1) ? acc[1] : acc[0];
    float s1 = (k & 1) ? acc[3] : acc[2];
    float s2 = (k & 1) ? acc[5] : acc[4];
    float s3 = (k & 1) ? acc[7] : acc[6];
    float t0 = (k & 2) ? s1 : s0;
    float t1 = (k & 2) ? s3 : s2;
    float a  = (k & 4) ? t1 : t0;
    float b  = __shfl_xor(a, 16, 32);
    const bool want_hi = (m & 8) != 0;
    const bool have_hi = (kh != 0);
    const float ssum = (want_hi == have_hi) ? a : b;

    const float inv = 1.0f / fmaxf(sqrtf(ssum), 1e-12f);

    float* op = out + (size_t)rowg * OUTW + (size_t)slice * DIM + 2 * kh;
#pragma unroll
    for (int c = 0; c < 16; ++c) {
        v2f t = y[c];
        t.x *= inv; t.y *= inv;
        __builtin_nontemporal_store(t, (v2f*)(op + 4 * c));
    }
}

// ---------------------------------------------------------------------------
extern "C" void kernel_launch(void* const* d_in, const int* in_sizes, int n_in,
                              void* d_out, int out_size, void* d_ws, size_t ws_size,
                              hipStream_t stream)
{
    const float* user = (const float*)d_in[0];
    const float* item = (const float*)d_in[1];
    const int*   hrow = (const int*)d_in[2];
    const int*   hcol = (const int*)d_in[3];
    const float* hval = (const float*)d_in[4];
    float* out = (float*)d_out;
    const int nnz = in_sizes[2];

    float* E     = (float*)d_ws;                 // 150000*64 f32 = 38.4 MB
    float* front = E + (size_t)NNODE * DIM;      // another 38.4 MB

    {
        int n = NNODE * (DIM / 4);
        init_concat<<<(n + 255) / 256, 256, 0, stream>>>(user, item, E, out);
    }

    for (int it = 0; it < 3; ++it) {
        hipMemsetAsync(front, 0, sizeof(float) * (size_t)NNODE * DIM, stream);

        // 2048 blocks x 8 waves = 16384 waves over 75000 chunks: ~4.6 chunks
        // per wave, enough for the ASYNCcnt double buffer to pipeline while
        // wave multiplexing hides the per-edge gather/atomic latency.
        spmm_scatter<<<2048, 256, 0, stream>>>(hrow, hcol, hval, E, front, nnz);

        int tiles   = NNODE / 16;                // 9375
        int nblocks = (tiles + 7) / 8;           // 8 tiles (waves) / block
        fuse_norm<<<nblocks, 256, 0, stream>>>(front, E, out, it + 1);
    }
}